// Clas_6957847020174
// MI455X (gfx1250) — compile-verified
//
#include <hip/hip_runtime.h>
#include <hip/hip_bf16.h>
#include <stdint.h>

// Problem constants (match the reference)
#define BATCH   512
#define NCROPS  5
#define TLEN    8192
#define NEGV    (-1e30f)

typedef __attribute__((ext_vector_type(2))) float v2f;
typedef __attribute__((ext_vector_type(8))) float v8f;

__device__ __forceinline__ uint32_t f32_to_ordered(float f) {
    uint32_t u = __float_as_uint(f);
    u ^= (uint32_t)((int32_t)u >> 31) | 0x80000000u;
    return u;
}
__device__ __forceinline__ float ordered_to_f32(uint32_t u) {
    uint32_t x = (u & 0x80000000u) ? (u ^ 0x80000000u) : ~u;
    return __uint_as_float(x);
}

// One workgroup per batch row. 256 threads = 8 wave32s.
__global__ __launch_bounds__(256)
void topk_bce_row_kernel(const float* __restrict__ scores,
                         const int*   __restrict__ label,
                         const int*   __restrict__ seqlen,
                         float*       __restrict__ row_loss)
{
    __shared__ uint32_t keys[TLEN];   // 32 KB: ordered keys of the masked crop-mean row
    __shared__ uint32_t hist[256];    // radix histogram
    __shared__ uint32_t sel[2];       // {selected digit, remaining rank}
    __shared__ float    s_sum;
    __shared__ uint32_t s_cnt;

    const int b    = blockIdx.x;
    const int tid  = threadIdx.x;
    const int wave = tid >> 5;
    const int lane = tid & 31;
    const int half = lane >> 4;       // 0: lanes 0-15, 1: lanes 16-31
    const int tt   = lane & 15;

    const int sl  = seqlen[b];
    const int lab = label[b];
    const size_t base = (size_t)b * NCROPS * TLEN;

    // ---- Phase A: crop mean via V_WMMA_F32_16X16X4_F32 (crops 0..3) + FMA (crop 4) ----
    // A[16x4] = 0.2 everywhere  ->  C[m,n] = 0.2 * sum_{c<4} scores[c, t0+n]  (all rows equal)
    v2f amat; amat.x = 0.2f; amat.y = 0.2f;

    // ISA B layout (4x16 f32): VGPR0 = {lanes0-15: K=0, lanes16-31: K=2},
    //                          VGPR1 = {lanes0-15: K=1, lanes16-31: K=3}
    // Two chunks (two WMMAs) per iteration: lanes 0-15 finalize chunk A (D row M=0),
    // lanes 16-31 finalize chunk B (D row M=8, identical value) -> zero divergence.
    const float* pB = scores + base + (size_t)(half ? 2 : 0) * TLEN + tt;
    const float* p4 = scores + base + (size_t)4 * TLEN + tt;

    for (int i = 0; i < 32; ++i) {
        const int t0A = (wave + 16 * i) * 16;  // chunk A
        const int t0B = t0A + 128;             // chunk B = chunk A + 8 chunks
        v2f bA, bB;
        bA.x = pB[t0A]; bA.y = pB[t0A + TLEN];   // crops 0/1 (lanes<16) or 2/3
        bB.x = pB[t0B]; bB.y = pB[t0B + TLEN];
        const int t0sel = half ? t0B : t0A;
        const float x4 = p4[t0sel];              // crop 4 for this lane's chunk

        v8f z = {};
        v8f dA = __builtin_amdgcn_wmma_f32_16x16x4_f32(
            false, amat, false, bA, (short)0, z, false, false);
        v8f dB = __builtin_amdgcn_wmma_f32_16x16x4_f32(
            false, amat, false, bB, (short)0, z, false, false);

        const float dd = half ? dB[0] : dA[0];   // mean*0.2 of crops 0..3 at column tt
        const float m  = dd + 0.2f * x4;
        const int   t  = t0sel + tt;
        const float v  = (t < sl) ? m : NEGV;
        keys[t] = f32_to_ordered(v);
    }
    __syncthreads();

    // ---- Phase B: exact top-k threshold via 4-pass MSB radix select on LDS keys ----
    const int k = (lab == 0) ? 1 : (sl / 16 + 1);   // k-1 < seqlen always (seqlen >= 16)

    uint32_t known = 0, kmask = 0;
    int krem = k;
    for (int round = 0; round < 4; ++round) {
        const int shift = 24 - 8 * round;

        hist[tid] = 0;
        __syncthreads();

        for (int i = tid; i < TLEN; i += 256) {
            uint32_t u = keys[i];
            if ((u & kmask) == known)
                atomicAdd(&hist[(u >> shift) & 255u], 1u);
        }
        __syncthreads();

        if (wave == 0) {
            // lane owns bins [lane*8, lane*8+8); suffix scan locates k-th largest's bucket
            uint32_t h[8], gsum = 0;
            #pragma unroll
            for (int j = 0; j < 8; ++j) { h[j] = hist[lane * 8 + j]; gsum += h[j]; }
            uint32_t s = gsum;
            #pragma unroll
            for (int off = 1; off < 32; off <<= 1) {
                uint32_t o = __shfl_down(s, off, 32);
                if (lane + off < 32) s += o;
            }
            uint32_t running = s - gsum;   // count of prefix-matching keys in higher bins
            #pragma unroll
            for (int j = 7; j >= 0; --j) {
                uint32_t c = h[j];
                if ((uint32_t)krem > running && (uint32_t)krem <= running + c) {
                    sel[0] = (uint32_t)(lane * 8 + j);
                    sel[1] = (uint32_t)krem - running;
                }
                running += c;
            }
        }
        __syncthreads();
        known |= sel[0] << shift;
        kmask |= (0xFFu << shift);
        krem = (int)sel[1];
        __syncthreads();
    }
    const uint32_t kthKey = known;   // exact key of the k-th largest element

    // ---- Phase C: sum of strictly-greater elements + tie fill (bit-exact vs sort) ----
    if (tid == 0) { s_sum = 0.0f; s_cnt = 0u; }
    __syncthreads();

    float    psum = 0.0f;
    uint32_t pcnt = 0u;
    for (int i = tid; i < TLEN; i += 256) {
        uint32_t u = keys[i];
        if (u > kthKey) { psum += ordered_to_f32(u); pcnt++; }
    }
    #pragma unroll
    for (int off = 16; off; off >>= 1) {
        psum += __shfl_down(psum, off, 32);
        pcnt += __shfl_down(pcnt, off, 32);
    }
    if (lane == 0) { atomicAdd(&s_sum, psum); atomicAdd(&s_cnt, pcnt); }
    __syncthreads();

    if (tid == 0) {
        const float kv  = ordered_to_f32(kthKey);
        const float tot = s_sum + (float)(k - (int)s_cnt) * kv;
        const float vl  = tot / (float)k;
        const float y   = (float)lab;
        // logaddexp(0, vl) - vl*y, numerically stable
        const float loss = fmaxf(vl, 0.0f) + log1pf(expf(-fabsf(vl))) - vl * y;
        row_loss[b] = loss;
    }
}

// Reduce 512 per-row losses -> mean scalar
__global__ __launch_bounds__(256)
void topk_bce_reduce_kernel(const float* __restrict__ row_loss, float* __restrict__ out)
{
    __shared__ float sh[8];
    const int tid  = threadIdx.x;
    const int lane = tid & 31;

    float v = row_loss[tid] + row_loss[tid + 256];
    #pragma unroll
    for (int off = 16; off; off >>= 1) v += __shfl_down(v, off, 32);
    if (lane == 0) sh[tid >> 5] = v;
    __syncthreads();
    if (tid < 8) {
        float x = sh[tid];
        x += __shfl_down(x, 4, 32);
        x += __shfl_down(x, 2, 32);
        x += __shfl_down(x, 1, 32);
        if (tid == 0) out[0] = x / (float)BATCH;
    }
}

extern "C" void kernel_launch(void* const* d_in, const int* in_sizes, int n_in,
                              void* d_out, int out_size, void* d_ws, size_t ws_size,
                              hipStream_t stream)
{
    (void)in_sizes; (void)n_in; (void)out_size; (void)ws_size;
    const float* scores = (const float*)d_in[0];
    const int*   label  = (const int*)d_in[1];
    const int*   seqlen = (const int*)d_in[2];
    float* out      = (float*)d_out;
    float* row_loss = (float*)d_ws;   // 512 floats, fully overwritten every call

    topk_bce_row_kernel<<<BATCH, 256, 0, stream>>>(scores, label, seqlen, row_loss);
    topk_bce_reduce_kernel<<<1, 256, 0, stream>>>(row_loss, out);
}